// RKPMLayer_12876311954301
// MI455X (gfx1250) — compile-verified
//
#include <hip/hip_runtime.h>
#include <math.h>

typedef __attribute__((ext_vector_type(2))) float v2f;
typedef __attribute__((ext_vector_type(4))) float v4f;
typedef __attribute__((ext_vector_type(8))) float v8f;

#define SUPPORT_FACTOR 2.5f
#define EPSF 1e-5f
#define MAXN 4096

// Branchless cubic B-spline, identical (in exact arithmetic) to the reference
// piecewise form: w = 4/3*( max(1-q,0)^3 - 4*max(0.5-q,0)^3 )
__device__ __forceinline__ float spline_w(float q) {
    float t1 = fmaxf(1.0f - q, 0.0f);
    float t2 = fmaxf(0.5f - q, 0.0f);
    float t1c = t1 * t1 * t1;
    float t2c = t2 * t2 * t2;
    return (4.0f / 3.0f) * (t1c - 4.0f * t2c);
}

__device__ __forceinline__ float pair_w(float dx, float dy, float inv_dil) {
    float dist = __builtin_amdgcn_sqrtf(dx * dx + dy * dy + 1e-10f);
    return spline_w(dist * inv_dil);
}

// Branchless monomial selector (pure v_cndmask chain, no divergence)
__device__ __forceinline__ float monomial(int col, float X0, float X1) {
    float v = 1.0f;
    v = (col == 1) ? X0 : v;
    v = (col == 2) ? X1 : v;
    v = (col == 3) ? X0 * X0 : v;
    v = (col == 4) ? X0 * X1 : v;
    v = (col == 5) ? X1 * X1 : v;
    v = (col > 5) ? 0.0f : v;
    return v;
}

// ------------- Kernel A1: NN-distance partial sums (4 threads per node) --------
__global__ __launch_bounds__(256) void nn_partial(const float* __restrict__ nodes,
                                                  float* __restrict__ ws, int N) {
    __shared__ __align__(16) float snodes[MAXN * 2];
    __shared__ float red[256];
    __shared__ float sd[64];
    const int tid = threadIdx.x;
    for (int i = tid; i < N * 2; i += 256) snodes[i] = nodes[i];
    __syncthreads();

    const int t  = blockIdx.x * 256 + tid;
    const int i  = t >> 2;            // node index
    const int jq = t & 3;             // quarter of the j-range
    const int jn = N >> 2;
    const int j0 = jq * jn;
    const float xi0 = snodes[2 * i], xi1 = snodes[2 * i + 1];
    float md = 3.4e38f;
    for (int j = j0; j < j0 + jn; ++j) {
        float dx = xi0 - snodes[2 * j];
        float dy = xi1 - snodes[2 * j + 1];
        float d2 = dx * dx + dy * dy;
        md = (j == i) ? md : fminf(md, d2);
    }
    red[tid] = md;
    __syncthreads();
    if ((tid & 3) == 0) {
        float m = fminf(fminf(red[tid], red[tid + 1]), fminf(red[tid + 2], red[tid + 3]));
        sd[tid >> 2] = sqrtf(m);      // precise sqrt: matches reference path
    }
    __syncthreads();
    if (tid < 32) red[tid] = sd[tid] + sd[tid + 32];
    __syncthreads();
    for (int s = 16; s > 0; s >>= 1) {
        if (tid < s) red[tid] += red[tid + s];
        __syncthreads();
    }
    if (tid == 0) ws[blockIdx.x] = red[0];
}

// ------------- Kernel A2: finish dilation (fixed order, deterministic) ---------
__global__ void nn_finish(float* __restrict__ ws, int nblocks, int N) {
    if (threadIdx.x == 0 && blockIdx.x == 0) {
        float s = 0.0f;
        for (int i = 0; i < nblocks; ++i) s += ws[i];
        float dil = SUPPORT_FACTOR * (s / (float)N);
        ws[64] = dil;
        ws[65] = 1.0f / dil;
    }
}

// ------------- Kernel B: fused WMMA moments + 3x3 inverse + dense outputs ------
// Block = 256 threads (8 wave32), handles 16 query points.
__global__ __launch_bounds__(256) void rkpm_main(const float* __restrict__ x,
                                                 const float* __restrict__ nodes,
                                                 const float* __restrict__ ws,
                                                 float* __restrict__ out,
                                                 int N, int B) {
    __shared__ __align__(16) float snodes[MAXN * 2]; // 32 KB node cache
    __shared__ float Sp[8][16][16];                  // per-wave WMMA D tiles
    __shared__ float xq[16][2];
    __shared__ float Mi[16][12];

    const int tid  = threadIdx.x;
    const int lane = tid & 31;
    const int wave = tid >> 5;
    const int q0   = blockIdx.x * 16;

    for (int i = tid; i < N * 2; i += 256) snodes[i] = nodes[i];
    if (tid < 32) xq[tid >> 1][tid & 1] = x[q0 * 2 + tid];
    __syncthreads();

    const float inv_dil = ws[65];

    // ---- Pass 1: S[16 queries x 6 monomials] += W(16x4) x T(4x16) via WMMA ----
    // A (16x4 f32): lanes 0-15 -> row m; VGPR0={K0,K2}, VGPR1={K1,K3}.
    // B (4x16 f32): VGPR0={rows K0,K2}, VGPR1={rows K1,K3}; col = lane&15.
    v8f acc = {};
    const int   m     = lane & 15;
    const int   half  = lane >> 4;
    const int   rbase = half * 8;
    const float qx0   = xq[m][0];
    const float qx1   = xq[m][1];
    const int   nper  = N >> 3;          // nodes handled by this wave
    const int   base  = wave * nper + half * 2;
    const v4f*  sn4   = (const v4f*)snodes;

    for (int it = 0; it < nper; it += 4) {
        const int k0 = base + it;               // even -> 16B aligned pair load
        const v4f nn = sn4[k0 >> 1];            // nodes k0:(x,y), k0+1:(x,y)

        v2f a;                                  // weights W[m, k0], W[m, k0+1]
        a.x = pair_w(qx0 - nn.x, qx1 - nn.y, inv_dil);
        a.y = pair_w(qx0 - nn.z, qx1 - nn.w, inv_dil);
        v2f b;                                  // centered monomials of the nodes
        b.x = monomial(m, nn.x - 0.5f, nn.y - 0.5f);
        b.y = monomial(m, nn.z - 0.5f, nn.w - 0.5f);

        acc = __builtin_amdgcn_wmma_f32_16x16x4_f32(false, a, false, b,
                                                    (short)0, acc, false, false);
    }

#pragma unroll
    for (int r = 0; r < 8; ++r) Sp[wave][rbase + r][m] = acc[r];
    __syncthreads();

    // ---- Assemble M from moments, regularize, invert (threads 0..15) ----------
    if (tid < 16) {
        float S[6];
#pragma unroll
        for (int a2 = 0; a2 < 6; ++a2) {
            float s = 0.0f;
#pragma unroll
            for (int w = 0; w < 8; ++w) s += Sp[w][tid][a2];
            S[a2] = s;
        }
        const float xc0 = xq[tid][0] - 0.5f;
        const float xc1 = xq[tid][1] - 0.5f;
        const float M00 = S[0] + EPSF;
        const float M01 = xc0 * S[0] - S[1];
        const float M02 = xc1 * S[0] - S[2];
        const float M11 = xc0 * xc0 * S[0] - 2.0f * xc0 * S[1] + S[3] + EPSF;
        const float M12 = xc0 * xc1 * S[0] - xc0 * S[2] - xc1 * S[1] + S[4];
        const float M22 = xc1 * xc1 * S[0] - 2.0f * xc1 * S[2] + S[5] + EPSF;

        const float c00 = M11 * M22 - M12 * M12;
        const float c01 = M02 * M12 - M01 * M22;
        const float c02 = M01 * M12 - M02 * M11;
        const float c11 = M00 * M22 - M02 * M02;
        const float c12 = M01 * M02 - M00 * M12;
        const float c22 = M00 * M11 - M01 * M01;
        const float det = M00 * c00 + M01 * c01 + M02 * c02;
        const float id  = 1.0f / det;

        Mi[tid][0] = c00 * id; Mi[tid][1] = c01 * id; Mi[tid][2] = c02 * id;
        Mi[tid][3] = c01 * id; Mi[tid][4] = c11 * id; Mi[tid][5] = c12 * id;
        Mi[tid][6] = c02 * id; Mi[tid][7] = c12 * id; Mi[tid][8] = c22 * id;
    }
    __syncthreads();

    // ---- Pass 2: dense phi / phi_x / phi_y, b128 nontemporal stores ------------
    const size_t BN = (size_t)B * (size_t)N;
#pragma unroll 1
    for (int qi = 0; qi < 2; ++qi) {
        const int   qm  = wave * 2 + qi;
        const int   bq  = q0 + qm;
        const float px0 = xq[qm][0], px1 = xq[qm][1];
        const float m0 = Mi[qm][0], m1 = Mi[qm][1], m2 = Mi[qm][2];
        const float m3 = Mi[qm][3], m4 = Mi[qm][4], m5 = Mi[qm][5];
        const float m6 = Mi[qm][6], m7 = Mi[qm][7], m8 = Mi[qm][8];
        float* __restrict__ row0 = out + (size_t)bq * (size_t)N;
        float* __restrict__ row1 = row0 + BN;
        float* __restrict__ row2 = row0 + 2 * BN;

        for (int n0 = lane * 4; n0 < N; n0 += 128) {
            const v4f pa = sn4[(n0 >> 1) + 0];   // nodes n0, n0+1
            const v4f pb = sn4[(n0 >> 1) + 1];   // nodes n0+2, n0+3
            const float X0[4] = {pa.x, pa.z, pb.x, pb.z};
            const float X1[4] = {pa.y, pa.w, pb.y, pb.w};
            float r0[4], r1[4], r2[4];
#pragma unroll
            for (int j = 0; j < 4; ++j) {
                const float dx = px0 - X0[j];
                const float dy = px1 - X1[j];
                const float dist = __builtin_amdgcn_sqrtf(dx * dx + dy * dy + 1e-10f);
                const float w = spline_w(dist * inv_dil);
                r0[j] =  w * (m0 + m1 * dx + m2 * dy);   // phi
                r1[j] = -w * (m3 + m4 * dx + m5 * dy);   // phi_x
                r2[j] = -w * (m6 + m7 * dx + m8 * dy);   // phi_y
            }
            v4f o0 = {r0[0], r0[1], r0[2], r0[3]};
            v4f o1 = {r1[0], r1[1], r1[2], r1[3]};
            v4f o2 = {r2[0], r2[1], r2[2], r2[3]};
            __builtin_nontemporal_store(o0, (v4f*)(row0 + n0));
            __builtin_nontemporal_store(o1, (v4f*)(row1 + n0));
            __builtin_nontemporal_store(o2, (v4f*)(row2 + n0));
        }
    }
}

extern "C" void kernel_launch(void* const* d_in, const int* in_sizes, int n_in,
                              void* d_out, int out_size, void* d_ws, size_t ws_size,
                              hipStream_t stream) {
    const float* x     = (const float*)d_in[0];
    const float* nodes = (const float*)d_in[1];
    float*       out   = (float*)d_out;
    float*       ws    = (float*)d_ws;

    const int B = in_sizes[0] / 2;  // 4096 query points
    const int N = in_sizes[1] / 2;  // 4096 nodes

    const int nnBlocks = (4 * N) / 256;   // 64
    nn_partial<<<nnBlocks, 256, 0, stream>>>(nodes, ws, N);
    nn_finish<<<1, 32, 0, stream>>>(ws, nnBlocks, N);
    rkpm_main<<<B / 16, 256, 0, stream>>>(x, nodes, ws, out, N, B);
}